// bahdanauattention_79173427134835
// MI455X (gfx1250) — compile-verified
//
#include <hip/hip_runtime.h>
#include <hip/hip_bf16.h>

typedef __attribute__((ext_vector_type(2))) float v2f;
typedef __attribute__((ext_vector_type(8))) float v8f;

#define BSZ 4
#define TSZ 512
#define DSZ 512

__device__ __forceinline__ float fast_tanh(float x) {
#if __has_builtin(__builtin_amdgcn_tanhf)
    return __builtin_amdgcn_tanhf(x);          // gfx1250 v_tanh_f32
#else
    float ax = __builtin_fabsf(x);
    float e  = __builtin_amdgcn_exp2f(ax * 2.885390081777927f);   // e^(2|x|)
    float r  = 1.0f - 2.0f * __builtin_amdgcn_rcpf(e + 1.0f);
    return x < 0.0f ? -r : r;
#endif
}

__device__ __forceinline__ float fast_exp(float x) {
    return __builtin_amdgcn_exp2f(x * 1.4426950408889634f);
}

// ---------------------------------------------------------------------------
// Kernel 1: Q = H @ Wq, K = H @ Wk   (M = B*T = 2048, N = K = 512), f32 WMMA.
// One wave -> one 16x16 tile of Q and of K (shared A operand loads).
// A layout (16x4 f32): lane m=lane&15, khalf=lane>>4; a[j] = A[m][k0+2*kh+j]
// B layout (4x16 f32): lane n=lane&15, khalf=lane>>4; b[j] = B[k0+2*kh+j][n]
// C layout: VGPR i -> row (i + 8*khalf), col = lane&15
// ---------------------------------------------------------------------------
__global__ __launch_bounds__(128) void qk_proj_kernel(
    const float* __restrict__ H, const float* __restrict__ Wq,
    const float* __restrict__ Wk, float* __restrict__ Q, float* __restrict__ K)
{
    const int lane = threadIdx.x & 31;
    const int wave = threadIdx.x >> 5;
    const int tile = blockIdx.x * 4 + wave;   // 4096 tiles = (2048/16)*(512/16)
    const int rowTile = tile >> 5;            // 0..127
    const int colTile = tile & 31;            // 0..31
    const int m  = lane & 15;
    const int kh = lane >> 4;

    const float* hrow = H + (size_t)(rowTile * 16 + m) * DSZ;
    const int n = colTile * 16 + m;           // B-operand column

    v8f accQ = {};
    v8f accK = {};
    for (int k0 = 0; k0 < DSZ; k0 += 4) {
        const int ka = k0 + 2 * kh;
        v2f a;  a.x  = hrow[ka];                    a.y  = hrow[ka + 1];
        v2f bq; bq.x = Wq[(size_t)ka * DSZ + n];    bq.y = Wq[(size_t)(ka + 1) * DSZ + n];
        v2f bk; bk.x = Wk[(size_t)ka * DSZ + n];    bk.y = Wk[(size_t)(ka + 1) * DSZ + n];
        accQ = __builtin_amdgcn_wmma_f32_16x16x4_f32(false, a, false, bq,
                                                     (short)0, accQ, false, false);
        accK = __builtin_amdgcn_wmma_f32_16x16x4_f32(false, a, false, bk,
                                                     (short)0, accK, false, false);
    }

    const int cr0 = rowTile * 16 + kh * 8;
    const int cc  = colTile * 16 + m;
#pragma unroll
    for (int i = 0; i < 8; i++) {
        Q[(size_t)(cr0 + i) * DSZ + cc] = accQ[i];
        K[(size_t)(cr0 + i) * DSZ + cc] = accK[i];
    }
}

// ---------------------------------------------------------------------------
// Kernel 2: energy[b,t,s] = sum_d v[d]*tanh(Q[b,t,d] + K[b,s,d])
// One wave per (t,s) pair, lanes strided over d (coalesced), shuffle-reduce.
// Grid: B * (T/16 t-tiles) * (T/128 s-chunks) = 512 blocks of 256 threads.
// ---------------------------------------------------------------------------
__global__ __launch_bounds__(256) void energy_kernel(
    const float* __restrict__ Q, const float* __restrict__ Kmat,
    const float* __restrict__ v, float* __restrict__ energy)
{
    const int blk    = blockIdx.x;
    const int sChunk = blk & 3;          // 4 chunks of 128 s
    const int tTile  = (blk >> 2) & 31;  // 32 tiles of 16 t
    const int b      = blk >> 7;
    const int t0     = tTile * 16;
    const int s0     = sChunk * 128;

    const float* Qb = Q    + ((size_t)b * TSZ + t0) * DSZ;
    const float* Kb = Kmat + (size_t)b * TSZ * DSZ;
    float*       Eb = energy + ((size_t)b * TSZ + t0) * TSZ;

    const int wave = threadIdx.x >> 5;
    const int lane = threadIdx.x & 31;

    for (int p = wave; p < 16 * 128; p += 8) {
        const int t = p & 15;
        const int s = s0 + (p >> 4);
        const float* qrow = Qb + (size_t)t * DSZ;
        const float* krow = Kb + (size_t)s * DSZ;

        float acc = 0.0f;
        for (int d = lane; d < DSZ; d += 32)
            acc = __builtin_fmaf(v[d], fast_tanh(qrow[d] + krow[d]), acc);

        for (int off = 16; off; off >>= 1)
            acc += __shfl_xor(acc, off, 32);
        if (lane == 0)
            Eb[(size_t)t * TSZ + s] = acc;
    }
}

// ---------------------------------------------------------------------------
// Kernel 3: in-place masked softmax of each (b,t) row of the weights buffer.
// One wave per row; lane holds 16 strided elements. All-masked rows -> 0.
// ---------------------------------------------------------------------------
__global__ __launch_bounds__(256) void softmax_kernel(
    float* __restrict__ W, const unsigned char* __restrict__ mask)
{
    const int wave = threadIdx.x >> 5;
    const int lane = threadIdx.x & 31;
    const int row  = blockIdx.x * 8 + wave;     // 0..2047
    const int b    = row >> 9;

    float* w = W + (size_t)row * TSZ;
    const unsigned char* mrow = mask + (size_t)b * TSZ;

    const float NEG_INF = -__builtin_inff();
    float vals[16];
    float vmax = NEG_INF;
#pragma unroll
    for (int i = 0; i < 16; i++) {
        const int s = lane + i * 32;
        const float e = w[s];
        vals[i] = (mrow[s] != 0) ? e : NEG_INF;
        vmax = __builtin_fmaxf(vmax, vals[i]);
    }
    for (int off = 16; off; off >>= 1)
        vmax = __builtin_fmaxf(vmax, __shfl_xor(vmax, off, 32));

    float sum = 0.0f;
#pragma unroll
    for (int i = 0; i < 16; i++) {
        const float p = (vals[i] == NEG_INF) ? 0.0f : fast_exp(vals[i] - vmax);
        vals[i] = p;
        sum += p;
    }
    for (int off = 16; off; off >>= 1)
        sum += __shfl_xor(sum, off, 32);

    const float inv = (sum > 0.0f) ? __builtin_amdgcn_rcpf(sum) : 0.0f;
#pragma unroll
    for (int i = 0; i < 16; i++)
        w[lane + i * 32] = vals[i] * inv;
}

// ---------------------------------------------------------------------------
// Kernel 4: context[b] = weights[b] (T x T) @ H[b] (T x D), f32 WMMA.
// ---------------------------------------------------------------------------
__global__ __launch_bounds__(128) void context_kernel(
    const float* __restrict__ W, const float* __restrict__ H,
    float* __restrict__ Cout)
{
    const int lane = threadIdx.x & 31;
    const int wave = threadIdx.x >> 5;
    const int g    = blockIdx.x * 4 + wave;  // 4096 tiles = 4 * 32 * 32
    const int b    = g >> 10;
    const int rem  = g & 1023;
    const int rT   = rem >> 5;
    const int cT   = rem & 31;

    const float* A  = W + (size_t)b * TSZ * TSZ;
    const float* Bm = H + (size_t)b * TSZ * DSZ;

    const int m  = lane & 15;
    const int kh = lane >> 4;
    const float* arow = A + (size_t)(rT * 16 + m) * TSZ;
    const int n = cT * 16 + m;

    v8f acc = {};
    for (int k0 = 0; k0 < TSZ; k0 += 4) {
        const int ka = k0 + 2 * kh;
        v2f a;  a.x  = arow[ka];                  a.y  = arow[ka + 1];
        v2f bb; bb.x = Bm[(size_t)ka * DSZ + n];  bb.y = Bm[(size_t)(ka + 1) * DSZ + n];
        acc = __builtin_amdgcn_wmma_f32_16x16x4_f32(false, a, false, bb,
                                                    (short)0, acc, false, false);
    }

    const int cr0 = rT * 16 + kh * 8;
    const int cc  = cT * 16 + m;
    float* Cb = Cout + (size_t)b * TSZ * DSZ;
#pragma unroll
    for (int i = 0; i < 8; i++)
        Cb[(size_t)(cr0 + i) * DSZ + cc] = acc[i];
}

// ---------------------------------------------------------------------------
extern "C" void kernel_launch(void* const* d_in, const int* in_sizes, int n_in,
                              void* d_out, int out_size, void* d_ws, size_t ws_size,
                              hipStream_t stream)
{
    const float*         Hh   = (const float*)d_in[0];          // (B,T,D)
    const float*         Wq   = (const float*)d_in[1];          // (D,D)
    const float*         Wk   = (const float*)d_in[2];          // (D,D)
    const float*         v    = (const float*)d_in[3];          // (D,)
    const unsigned char* mask = (const unsigned char*)d_in[4];  // (B,T) bool

    float* out     = (float*)d_out;
    float* context = out;                                   // (B,T,D)
    float* weights = out + (size_t)BSZ * TSZ * DSZ;         // (B,T,T)

    float* Q = (float*)d_ws;                                // (B*T, D)
    float* K = Q + (size_t)BSZ * TSZ * DSZ;                 // (B*T, D)

    qk_proj_kernel<<<1024, 128, 0, stream>>>(Hh, Wq, Wk, Q, K);
    energy_kernel<<<512, 256, 0, stream>>>(Q, K, v, weights);
    softmax_kernel<<<256, 256, 0, stream>>>(weights, mask);
    context_kernel<<<1024, 128, 0, stream>>>(weights, Hh, context);
}